// I24DualEMA_19490561590093
// MI455X (gfx1250) — compile-verified
//
#include <hip/hip_runtime.h>
#include <stdint.h>

// ---------------------------------------------------------------------------
// Dual quantized EMA scan, MI455X (gfx1250).
//
// Roofline: 268 MB read + 134 MB write minimum -> ~17 us @ 23.3 TB/s.
// Strategy: chunk the T axis (CHUNK=512) with a WARM=384 zero-state warm-up
// (alpha^384 ~ 2.8e-9 << 2^-23, state reconverges below the q24 LSB), giving
// 256 blocks x 256 threads = 2048 waves. HBM latency is hidden by the CDNA5
// Tensor Data Mover: double-buffered tensor_load_to_lds tiles
// (16 timesteps x 256 ch x 4B = 16 KB per stream), s_wait_tensorcnt + barriers
// for the pipeline, sequential quantized recurrence out of LDS.
// ---------------------------------------------------------------------------

typedef unsigned int v4u __attribute__((ext_vector_type(4)));
typedef int          v8i __attribute__((ext_vector_type(8)));
typedef int          v4i __attribute__((ext_vector_type(4)));

#define EMA_ALPHA 0.95f
#define NB   16      // batch
#define NT   8192    // time
#define NC   256     // channels
#define CHUNK 512    // output timesteps per block
#define WARM  384    // warm-up timesteps (alpha^384 << 2^-23)
#define TILE  16     // timesteps per LDS tile

__device__ __forceinline__ float q24(float v) {
    float r = rintf(v * 8388608.f);                    // v_rndne_f32 (RNE, matches jnp.round)
    r = fminf(fmaxf(r, -8388608.f), 8388607.f);
    return r * (1.f / 8388608.f);
}
__device__ __forceinline__ float q16(float v) {
    float r = rintf(v * 32768.f);
    r = fminf(fmaxf(r, -32768.f), 32767.f);
    return r * (1.f / 32768.f);
}

// Issue one TDM load of a [TILE x NC] fp32 tile (row-contiguous) into LDS.
// D# built per CDNA5 ISA ch.8: group0 {count=1, lds_addr, global_addr, type=2},
// group1 {data_size=4B, tensor_dim0=NC, tensor_dim1=TILE, tile=NCxTILE,
//         dim0_stride=NC}.  Groups 2/3 zero (2D tensor).
__device__ __forceinline__ void tdm_load_tile(const float* gptr, uint32_t lds_off)
{
    uint64_t ga    = (uint64_t)(uintptr_t)gptr;
    uint32_t ga_lo = __builtin_amdgcn_readfirstlane((uint32_t)ga);
    uint32_t ga_hi = __builtin_amdgcn_readfirstlane((uint32_t)(ga >> 32));
    uint32_t loff  = __builtin_amdgcn_readfirstlane(lds_off);

    v4u g0;
    g0.x = 1u;                                    // count=1 (valid user descriptor)
    g0.y = loff;                                  // lds_addr (bytes)
    g0.z = ga_lo;                                 // global_addr[31:0]
    g0.w = (ga_hi & 0x01FFFFFFu) | 0x80000000u;   // global_addr[56:32] | type=2<<30

    v8i g1;
    g1[0] = 0x00020000;            // workgroup_mask=0, data_size=2 (4 bytes)
    g1[1] = (int)(NC << 16);       // tensor_dim0[15:0]=256 (atomic_barrier_addr=0)
    g1[2] = (int)(TILE << 16);     // tensor_dim0[31:16]=0, tensor_dim1[15:0]=TILE
    g1[3] = (int)(NC << 16);       // tensor_dim1[31:16]=0, tile_dim0=256
    g1[4] = TILE;                  // tile_dim1=TILE, tile_dim2=0
    g1[5] = NC;                    // tensor_dim0_stride = 256 elements
    g1[6] = 0;                     // stride hi / dim1_stride lo
    g1[7] = 0;

    v4i z4 = (v4i)0;
    v8i z8 = (v8i)0;
    __builtin_amdgcn_tensor_load_to_lds(g0, g1, z4, z4, z8, 0);
}

__global__ void __launch_bounds__(256)
ema_scan_kernel(const float* __restrict__ x, const float* __restrict__ y,
                float* __restrict__ out)
{
    // [buf][stream][TILE*NC] : 2 * 2 * 16KB = 64 KB of the 320 KB WGP LDS
    __shared__ float lds[2][2][TILE * NC];

    const int c     = threadIdx.x;        // channel 0..255
    const int b     = blockIdx.x;         // batch 0..15
    const int chunk = blockIdx.y;         // 0..15

    const int t_out0  = chunk * CHUNK;
    const int t_begin = (chunk == 0) ? 0 : (t_out0 - WARM);
    const int ntiles  = (t_out0 + CHUNK - t_begin) / TILE;

    const float* xb = x + ((size_t)b * NT + t_begin) * NC;
    const float* yb = y + ((size_t)b * NT + t_begin) * NC;
    float*       ob = out + (size_t)b * NT * NC + c;

    const uint32_t lds_base = (uint32_t)(uintptr_t)&lds[0][0][0];
    const bool issuer = (threadIdx.x < 32);   // wave 0 drives the TDM

    if (issuer) {
        tdm_load_tile(xb, lds_base + 0u);
        tdm_load_tile(yb, lds_base + (uint32_t)(TILE * NC * 4));
    }

    float s0 = 0.f, s1 = 0.f;
    int t = t_begin;

    for (int i = 0; i < ntiles; ++i) {
        const int buf = i & 1;
        if (issuer) {
            if (i + 1 < ntiles) {
                const int nb = (i + 1) & 1;
                tdm_load_tile(xb + (size_t)(i + 1) * TILE * NC,
                              lds_base + (uint32_t)((nb * 2 + 0) * TILE * NC * 4));
                tdm_load_tile(yb + (size_t)(i + 1) * TILE * NC,
                              lds_base + (uint32_t)((nb * 2 + 1) * TILE * NC * 4));
                __builtin_amdgcn_s_wait_tensorcnt(2);  // tile i complete (in-order)
            } else {
                __builtin_amdgcn_s_wait_tensorcnt(0);
            }
        }
        __syncthreads();                               // tile i visible to all waves

        const float* lx = &lds[buf][0][c];
        const float* ly = &lds[buf][1][c];
        const bool emit = (t >= t_out0);               // tile-uniform (WARM,CHUNK % TILE == 0)

        #pragma unroll
        for (int k = 0; k < TILE; ++k) {
            float xv = lx[k * NC];
            float yv = ly[k * NC];
            s0 = q24(fmaf(EMA_ALPHA, s0, (1.f - EMA_ALPHA) * xv));
            s1 = q24(fmaf(EMA_ALPHA, s1, (1.f - EMA_ALPHA) * yv));
            if (emit)
                ob[(size_t)(t + k) * NC] = q16(s0) + q16(s1);
        }
        t += TILE;
        __syncthreads();                               // done reading buf before TDM reuses it
    }
}

extern "C" void kernel_launch(void* const* d_in, const int* in_sizes, int n_in,
                              void* d_out, int out_size, void* d_ws, size_t ws_size,
                              hipStream_t stream)
{
    const float* x = (const float*)d_in[0];
    const float* y = (const float*)d_in[1];
    float* out     = (float*)d_out;

    dim3 grid(NB, NT / CHUNK);   // 16 x 16 = 256 blocks, 256 threads (8 waves) each
    ema_scan_kernel<<<grid, 256, 0, stream>>>(x, y, out);
}